// GINConvModule_13520557048111
// MI455X (gfx1250) — compile-verified
//
#include <hip/hip_runtime.h>
#include <hip/hip_bf16.h>

typedef float v2f __attribute__((ext_vector_type(2)));
typedef float v8f __attribute__((ext_vector_type(8)));

#define N_NODES 50000
#define F_IN    128
#define H_DIM   64
#define HP      68          // padded LDS stride for h1 tile (bank-conflict avoidance)
#define BN_EPS  1e-5f

// ---------------------------------------------------------------------------
// Kernel 1: h0 = x  (GIN eps=0 -> (1+eps)*x + agg == x + agg), zero stats
// ---------------------------------------------------------------------------
__global__ void k_init(const float* __restrict__ x, float* __restrict__ h0,
                       float* __restrict__ sums, float* __restrict__ sumsq) {
    long long idx = (long long)blockIdx.x * blockDim.x + threadIdx.x;
    const long long total = (long long)N_NODES * F_IN;
    if (idx < total) h0[idx] = x[idx];
    if (idx < H_DIM) { sums[idx] = 0.0f; sumsq[idx] = 0.0f; }
}

// ---------------------------------------------------------------------------
// Kernel 2: scatter-add  h0[dst] += x[src]   (32 lanes/edge, float4 per lane)
// ---------------------------------------------------------------------------
__global__ void k_scatter(const float* __restrict__ x,
                          const long long* __restrict__ ei,
                          float* __restrict__ h0, int E) {
    long long t = (long long)blockIdx.x * blockDim.x + threadIdx.x;
    long long e = t >> 5;
    int j = (int)(t & 31);                // which group of 4 floats (4*32 = 128)
    if (e >= E) return;
    long long src = ei[e];
    long long dst = ei[(long long)E + e];
    const float4 v = *(const float4*)(x + src * F_IN + 4 * j);
    float* p = h0 + dst * F_IN + 4 * j;
    atomicAdd(p + 0, v.x);
    atomicAdd(p + 1, v.y);
    atomicAdd(p + 2, v.z);
    atomicAdd(p + 3, v.w);
}

// ---------------------------------------------------------------------------
// Kernel 3: fused MLP  h2 = relu(relu(h0@W1+b1)@W2+b2)  via V_WMMA_F32_16X16X4_F32
//           + per-column sum / sumsq accumulation for BatchNorm.
// 128 threads = 4 waves; each wave owns a 16-row tile. Out-of-range rows are
// CLAMPED to N-1 for loads (branch-free K-loop, EXEC all-ones at every WMMA);
// the epilogue masks stores and statistics by the true row index.
// ---------------------------------------------------------------------------
__global__ __launch_bounds__(128)
void k_mlp(const float* __restrict__ h0, const float* __restrict__ W1,
           const float* __restrict__ b1, const float* __restrict__ W2,
           const float* __restrict__ b2, float* __restrict__ h2,
           float* __restrict__ sums, float* __restrict__ sumsq) {
    __shared__ float sW1[F_IN * H_DIM];     // 32 KB
    __shared__ float sB1[H_DIM];
    __shared__ float sB2[H_DIM];
    __shared__ float sH1[4][16 * HP];       // ~17 KB, padded rows

    const int tid  = threadIdx.x;
    const int wave = tid >> 5;
    const int lane = tid & 31;
    const int m    = lane & 15;             // M index for A, N index for B/C/D
    const int kh   = lane >> 4;             // half: K pair select (A/B), M+8 (C/D)

    for (int i = tid; i < F_IN * H_DIM; i += 128) sW1[i] = W1[i];
    if (tid < H_DIM) { sB1[tid] = b1[tid]; sB2[tid] = b2[tid]; }
    __syncthreads();

    const int rowBase  = (blockIdx.x * 4 + wave) * 16;
    const int row      = rowBase + m;
    const int rowClamp = row < N_NODES ? row : (N_NODES - 1);   // branch-free A loads
    const float* aRow  = h0 + (long long)rowClamp * F_IN + kh * 2;

    // ---------------- GEMM1: [16x128] @ [128x64] ----------------
    v8f acc[4];
#pragma unroll
    for (int nt = 0; nt < 4; ++nt)
        acc[nt] = (v8f){0.f,0.f,0.f,0.f,0.f,0.f,0.f,0.f};

    for (int kk = 0; kk < F_IN / 4; ++kk) {
        const int k0 = kk * 4 + kh * 2;
        const v2f a = *(const v2f*)(aRow + kk * 4);     // global_load_b64
#pragma unroll
        for (int nt = 0; nt < 4; ++nt) {
            v2f b;
            b[0] = sW1[(k0)     * H_DIM + nt * 16 + m];
            b[1] = sW1[(k0 + 1) * H_DIM + nt * 16 + m];
            acc[nt] = __builtin_amdgcn_wmma_f32_16x16x4_f32(
                false, a, false, b, (short)0, acc[nt], false, false);
        }
    }

    // bias + relu -> LDS tile (A operand for GEMM2)
#pragma unroll
    for (int nt = 0; nt < 4; ++nt) {
#pragma unroll
        for (int r = 0; r < 8; ++r) {
            const int mm  = r + 8 * kh;
            const int col = nt * 16 + m;
            float v = acc[nt][r] + sB1[col];
            sH1[wave][mm * HP + col] = v > 0.f ? v : 0.f;
        }
    }
    __syncthreads();

    // ---------------- GEMM2: [16x64] @ [64x64] ----------------
    v8f acc2[4];
#pragma unroll
    for (int nt = 0; nt < 4; ++nt)
        acc2[nt] = (v8f){0.f,0.f,0.f,0.f,0.f,0.f,0.f,0.f};

    for (int kk = 0; kk < H_DIM / 4; ++kk) {
        const int k0 = kk * 4 + kh * 2;
        v2f a;
        a[0] = sH1[wave][m * HP + k0];
        a[1] = sH1[wave][m * HP + k0 + 1];
#pragma unroll
        for (int nt = 0; nt < 4; ++nt) {
            v2f b;
            b[0] = W2[(k0)     * H_DIM + nt * 16 + m];
            b[1] = W2[(k0 + 1) * H_DIM + nt * 16 + m];
            acc2[nt] = __builtin_amdgcn_wmma_f32_16x16x4_f32(
                false, a, false, b, (short)0, acc2[nt], false, false);
        }
    }

    // bias + relu -> h2, and accumulate column sums / sums of squares
#pragma unroll
    for (int nt = 0; nt < 4; ++nt) {
        float s = 0.f, q = 0.f;
#pragma unroll
        for (int r = 0; r < 8; ++r) {
            const int mm   = r + 8 * kh;
            const int grow = rowBase + mm;
            const int col  = nt * 16 + m;
            float v = acc2[nt][r] + sB2[col];
            v = v > 0.f ? v : 0.f;
            if (grow < N_NODES) {
                h2[(long long)grow * H_DIM + col] = v;
                s += v;
                q += v * v;
            }
        }
        // combine lane L with lane L^16 (same column, other 8 rows) — wave32
        s += __shfl_xor(s, 16, 32);
        q += __shfl_xor(q, 16, 32);
        if (lane < 16) {
            atomicAdd(&sums[nt * 16 + lane], s);
            atomicAdd(&sumsq[nt * 16 + lane], q);
        }
    }
}

// ---------------------------------------------------------------------------
// Kernel 4: batch-norm finalize (training-mode batch stats, biased variance)
// ---------------------------------------------------------------------------
__global__ void k_norm(const float* __restrict__ h2, const float* __restrict__ sums,
                       const float* __restrict__ sumsq, const float* __restrict__ gamma,
                       const float* __restrict__ beta, float* __restrict__ out) {
    long long idx = (long long)blockIdx.x * blockDim.x + threadIdx.x;
    if (idx >= (long long)N_NODES * H_DIM) return;
    const int c = (int)(idx & (H_DIM - 1));
    const float invN = 1.0f / (float)N_NODES;
    const float mean = sums[c] * invN;
    const float var  = sumsq[c] * invN - mean * mean;
    const float inv  = rsqrtf(var + BN_EPS);
    out[idx] = gamma[c] * (h2[idx] - mean) * inv + beta[c];
}

// ---------------------------------------------------------------------------
extern "C" void kernel_launch(void* const* d_in, const int* in_sizes, int n_in,
                              void* d_out, int out_size, void* d_ws, size_t ws_size,
                              hipStream_t stream) {
    const float*     x     = (const float*)d_in[0];
    const long long* ei    = (const long long*)d_in[1];   // int64 edge_index [2,E]
    const float*     W1    = (const float*)d_in[2];
    const float*     b1    = (const float*)d_in[3];
    const float*     W2    = (const float*)d_in[4];
    const float*     b2    = (const float*)d_in[5];
    const float*     gamma = (const float*)d_in[6];
    const float*     beta  = (const float*)d_in[7];
    float*           out   = (float*)d_out;

    const int E = in_sizes[1] / 2;

    // workspace layout: h0 [N*128] | h2 [N*64] | sums[64] | sumsq[64]
    char*  ws    = (char*)d_ws;
    float* h0    = (float*)ws;
    float* h2    = (float*)(ws + (size_t)N_NODES * F_IN * sizeof(float));
    float* sums  = (float*)(ws + (size_t)N_NODES * (F_IN + H_DIM) * sizeof(float));
    float* sumsq = sums + H_DIM;

    {
        const long long total = (long long)N_NODES * F_IN;
        k_init<<<(int)((total + 255) / 256), 256, 0, stream>>>(x, h0, sums, sumsq);
    }
    {
        const long long total = (long long)E * 32;
        k_scatter<<<(int)((total + 255) / 256), 256, 0, stream>>>(x, ei, h0, E);
    }
    {
        const int blocks = (N_NODES + 63) / 64;   // 4 waves x 16 rows per block
        k_mlp<<<blocks, 128, 0, stream>>>(h0, W1, b1, W2, b2, h2, sums, sumsq);
    }
    {
        const long long total = (long long)N_NODES * H_DIM;
        k_norm<<<(int)((total + 255) / 256), 256, 0, stream>>>(h2, sums, sumsq, gamma, beta, out);
    }
}